// Attention_12008728560084
// MI455X (gfx1250) — compile-verified
//
#include <hip/hip_runtime.h>
#include <hip/hip_bf16.h>
#include <math.h>

#define BB 32
#define SS 2048
#define HH 1024

typedef float v2f __attribute__((ext_vector_type(2)));
typedef float v8f __attribute__((ext_vector_type(8)));

// ---------------------------------------------------------------------------
// Small fp32 WMMA GEMM: C[32 x N] = A[32 x K] * B + bias, optional DyT epilogue.
// One wave (32 threads) per 16x16 output tile, K-loop of V_WMMA_F32_16X16X4_F32.
// TRANSB=true : logical B[k][n] = Bm[n*K + k]   (i.e. C = A @ Bm^T)
// TRANSB=false: logical B[k][n] = Bm[k*N + n]   (i.e. C = A @ Bm)
// ---------------------------------------------------------------------------
template <bool TRANSB, bool DYT>
__global__ __launch_bounds__(32) void wmma_gemm32(
    const float* __restrict__ A, const float* __restrict__ Bm,
    const float* __restrict__ bias, float* __restrict__ C,
    const float* __restrict__ gamma, const float* __restrict__ beta,
    const float* __restrict__ alpha, int N, int K) {
  const int lane = threadIdx.x & 31;
  const int half = lane >> 4;  // 0: lanes 0-15, 1: lanes 16-31
  const int l15  = lane & 15;
  const int n0 = blockIdx.x * 16;
  const int m0 = blockIdx.y * 16;

  const int koff = half * 2;            // A/B K-offset for this half-wave
  const float* Arow = A + (m0 + l15) * K;

  v8f c = {0.f, 0.f, 0.f, 0.f, 0.f, 0.f, 0.f, 0.f};

  for (int k0 = 0; k0 < K; k0 += 4) {
    v2f a, b;
    a.x = Arow[k0 + koff];
    a.y = Arow[k0 + koff + 1];
    if (TRANSB) {
      const float* Brow = Bm + (n0 + l15) * K;
      b.x = Brow[k0 + koff];
      b.y = Brow[k0 + koff + 1];
    } else {
      b.x = Bm[(size_t)(k0 + koff) * N + (n0 + l15)];
      b.y = Bm[(size_t)(k0 + koff + 1) * N + (n0 + l15)];
    }
    c = __builtin_amdgcn_wmma_f32_16x16x4_f32(false, a, false, b, (short)0, c,
                                              false, false);
  }

  const int n = n0 + l15;
  const float bia = bias ? bias[n] : 0.0f;
  float g = 0.f, bt = 0.f, al = 0.f;
  if (DYT) { g = gamma[n]; bt = beta[n]; al = alpha[0]; }
#pragma unroll
  for (int v = 0; v < 8; ++v) {
    const int m = m0 + v + half * 8;
    float val = c[v] + bia;
    if (DYT) val = g * tanhf(al * val) + bt;
    C[m * N + n] = val;
  }
}

// qbk[b] = dot(q[b,:], bk)
__global__ __launch_bounds__(256) void qbk_kernel(const float* __restrict__ q,
                                                  const float* __restrict__ bk,
                                                  float* __restrict__ qbk) {
  __shared__ float red[256];
  const int b = blockIdx.x, t = threadIdx.x;
  float acc = 0.f;
  for (int h = t; h < HH; h += 256) acc += q[b * HH + h] * bk[h];
  red[t] = acc;
  __syncthreads();
  for (int o = 128; o >= 1; o >>= 1) {
    if (t < o) red[t] += red[t + o];
    __syncthreads();
  }
  if (t == 0) qbk[b] = red[0];
}

// energy[b,s] = (enc[b,s,:] . qhat[b,:] + qbk[b]) / sqrt(H)
// grid: (S/64, B), block 256 = 8 waves, one row per wave per iteration.
__global__ __launch_bounds__(256) void energy_kernel(
    const float* __restrict__ enc, const float* __restrict__ qhat,
    const float* __restrict__ qbk, float* __restrict__ energy) {
  const int b = blockIdx.y;
  const int s0 = blockIdx.x * 64;
  const int lane = threadIdx.x & 31;
  const int wave = threadIdx.x >> 5;

  // preload qhat[b] slice: lane holds h = 4*(lane + 32*i), i < 8
  float4 qh[8];
  const float4* qh4 = (const float4*)(qhat + b * HH);
#pragma unroll
  for (int i = 0; i < 8; ++i) qh[i] = qh4[lane + 32 * i];
  const float qc = qbk[b];

  for (int r = wave; r < 64; r += 8) {
    const int s = s0 + r;
    const float4* e4 = (const float4*)(enc + ((size_t)b * SS + s) * HH);
    // prefetch the next row this wave will process (global_prefetch_b8)
    if (r + 8 < 64) {
      const float4* e4n = (const float4*)(enc + ((size_t)b * SS + s + 8) * HH);
      __builtin_prefetch(&e4n[lane], 0, 0);
    }
    float acc = 0.f;
#pragma unroll
    for (int i = 0; i < 8; ++i) {
      float4 ev = e4[lane + 32 * i];
      acc += ev.x * qh[i].x + ev.y * qh[i].y + ev.z * qh[i].z + ev.w * qh[i].w;
    }
#pragma unroll
    for (int off = 16; off >= 1; off >>= 1) acc += __shfl_xor(acc, off, 32);
    if (lane == 0) energy[b * SS + s] = (acc + qc) * 0.03125f;  // 1/sqrt(1024)
  }
}

// attention_weights[b,:] = softmax(energy[b,:]) over S
__global__ __launch_bounds__(256) void softmax_kernel(
    const float* __restrict__ energy, float* __restrict__ attn) {
  __shared__ float red[256];
  const int b = blockIdx.x, t = threadIdx.x;
  const float* e = energy + b * SS;
  float vals[SS / 256];
  float m = -INFINITY;
#pragma unroll
  for (int i = 0; i < SS / 256; ++i) {
    vals[i] = e[t + 256 * i];
    m = fmaxf(m, vals[i]);
  }
  red[t] = m;
  __syncthreads();
  for (int o = 128; o >= 1; o >>= 1) {
    if (t < o) red[t] = fmaxf(red[t], red[t + o]);
    __syncthreads();
  }
  m = red[0];
  __syncthreads();
  float sum = 0.f;
#pragma unroll
  for (int i = 0; i < SS / 256; ++i) {
    vals[i] = __expf(vals[i] - m);
    sum += vals[i];
  }
  red[t] = sum;
  __syncthreads();
  for (int o = 128; o >= 1; o >>= 1) {
    if (t < o) red[t] += red[t + o];
    __syncthreads();
  }
  const float inv = 1.0f / red[0];
#pragma unroll
  for (int i = 0; i < SS / 256; ++i) attn[b * SS + t + 256 * i] = vals[i] * inv;
}

// partial[b][sc][h] = sum_{s in chunk sc} attn[b,s] * enc[b,s,h]
// grid: (8 s-chunks, 32 batches), block 256; thread t owns columns 4t..4t+3,
// so each block streams fully contiguous 4 KB rows with b128 loads.
__global__ __launch_bounds__(256) void ctx_partial_kernel(
    const float* __restrict__ enc, const float* __restrict__ attn,
    float* __restrict__ partial) {
  const int sc = blockIdx.x, b = blockIdx.y;
  const int t = threadIdx.x;
  const float4* ebase =
      (const float4*)(enc + ((size_t)b * SS + sc * 256) * HH) + t;
  const float* w = attn + b * SS + sc * 256;
  float4 acc = {0.f, 0.f, 0.f, 0.f};
#pragma unroll 4
  for (int s = 0; s < 256; ++s) {
    const float4 ev = ebase[(size_t)s * (HH / 4)];
    const float ws_ = w[s];
    acc.x += ws_ * ev.x;
    acc.y += ws_ * ev.y;
    acc.z += ws_ * ev.z;
    acc.w += ws_ * ev.w;
  }
  ((float4*)(partial + (size_t)(b * 8 + sc) * HH))[t] = acc;
}

// ctx0[b,h] = sum_sc partial[b][sc][h]
__global__ __launch_bounds__(256) void ctx_reduce_kernel(
    const float* __restrict__ partial, float* __restrict__ ctx0) {
  const int idx = blockIdx.x * 256 + threadIdx.x;  // 0..32767
  const int b = idx >> 10, h = idx & (HH - 1);
  float acc = 0.f;
#pragma unroll
  for (int sc = 0; sc < 8; ++sc) acc += partial[(b * 8 + sc) * HH + h];
  ctx0[idx] = acc;
}

extern "C" void kernel_launch(void* const* d_in, const int* in_sizes, int n_in,
                              void* d_out, int out_size, void* d_ws,
                              size_t ws_size, hipStream_t stream) {
  (void)in_sizes; (void)n_in; (void)out_size; (void)ws_size;
  const float* x     = (const float*)d_in[0];   // [B,H]
  const float* enc   = (const float*)d_in[1];   // [B,S,H]
  const float* Wq    = (const float*)d_in[2];   // [H,H]
  const float* bq    = (const float*)d_in[3];   // [H]
  const float* Wk    = (const float*)d_in[4];   // [H,H]
  const float* bk    = (const float*)d_in[5];   // [H]
  const float* Wv    = (const float*)d_in[6];   // [H,H]
  const float* bv    = (const float*)d_in[7];   // [H]
  const float* alpha = (const float*)d_in[8];   // [1]
  const float* gamma = (const float*)d_in[9];   // [H]
  const float* beta  = (const float*)d_in[10];  // [H]

  float* ws      = (float*)d_ws;
  float* q       = ws;                 // 32768
  float* qhat    = ws + 32768;         // 32768
  float* qbk     = ws + 65536;         // 32 (padded to 64)
  float* energy  = ws + 65600;         // B*S = 65536
  float* partial = ws + 131136;        // 32*8*1024 = 262144
  float* ctx0    = ws + 393280;        // 32768   (total ~1.7 MB)

  float* out_ctx  = (float*)d_out;            // [B,H]
  float* out_attn = (float*)d_out + BB * HH;  // [B,S]

  // 1) q = x @ Wq^T + bq           (WMMA fp32)
  wmma_gemm32<true, false><<<dim3(HH / 16, BB / 16), 32, 0, stream>>>(
      x, Wq, bq, q, nullptr, nullptr, nullptr, HH, HH);
  // 2) qhat = q @ Wk               (WMMA fp32)
  wmma_gemm32<false, false><<<dim3(HH / 16, BB / 16), 32, 0, stream>>>(
      q, Wk, nullptr, qhat, nullptr, nullptr, nullptr, HH, HH);
  // 3) qbk[b] = q[b] . bk  (softmax-invariant constant, kept for exactness)
  qbk_kernel<<<BB, 256, 0, stream>>>(q, bk, qbk);
  // 4) energy: one streaming pass over enc (268 MB)
  energy_kernel<<<dim3(SS / 64, BB), 256, 0, stream>>>(enc, qhat, qbk, energy);
  // 5) softmax over S -> attention weights output
  softmax_kernel<<<BB, 256, 0, stream>>>(energy, out_attn);
  // 6) ctx0 = attn-weighted sum over enc: second pass (L2-resident per batch)
  ctx_partial_kernel<<<dim3(8, BB), 256, 0, stream>>>(enc, out_attn, partial);
  ctx_reduce_kernel<<<(BB * HH) / 256, 256, 0, stream>>>(partial, ctx0);
  // 7) context = DyT(ctx0 @ Wv^T + bv)   (WMMA fp32 + tanh epilogue)
  wmma_gemm32<true, true><<<dim3(HH / 16, BB / 16), 32, 0, stream>>>(
      ctx0, Wv, bv, out_ctx, gamma, beta, alpha, HH, HH);
}